// DecoderSplatting_13443247636892
// MI455X (gfx1250) — compile-verified
//
#include <hip/hip_runtime.h>

// ---------------- problem constants (fixed shapes from reference) -----------
#define BB 2
#define VV 4
#define CC 14
#define HH 512
#define WW 512
#define HWX (HH * WW)                 // 262144 pixels per (b,v) slice
#define NBV (BB * VV)                 // 8 slices
#define NPIX ((size_t)NBV * HWX)      // 2097152 total pixels

#define TPB 256
#define PPT 4                         // pixels per thread
#define PX_PER_BLOCK (TPB * PPT)      // 1024
#define BLOCKS_PER_BV (HWX / PX_PER_BLOCK)   // 256
#define NBLOCKS (NBV * BLOCKS_PER_BV)        // 2048

// output layout: means(3) | cov(9) | opacity(1) | rgb(3) | big | small
#define OFF_MEANS ((size_t)0)
#define OFF_COV   ((size_t)NPIX * 3)
#define OFF_OP    (OFF_COV + (size_t)NPIX * 9)
#define OFF_RGB   (OFF_OP + (size_t)NPIX)
#define OFF_BIG   (OFF_RGB + (size_t)NPIX * 3)
#define OFF_SMALL (OFF_BIG + 1)

#define L2E 1.44269504088896340736f
#define LN2 0.69314718055994530942f

// --------- fast math on hardware transcendental units (v_exp/log/rcp/rsq) ---
__device__ __forceinline__ float fexp2(float x)  { return __builtin_amdgcn_exp2f(x); }
__device__ __forceinline__ float flog2(float x)  { return __builtin_amdgcn_logf(x); }
__device__ __forceinline__ float frcp(float x)   { return __builtin_amdgcn_rcpf(x); }
__device__ __forceinline__ float frsq(float x)   { return __builtin_amdgcn_rsqf(x); }
__device__ __forceinline__ float fsqrt_(float x) { return __builtin_amdgcn_sqrtf(x); }
__device__ __forceinline__ float flog_(float x)  { return flog2(x) * LN2; }
__device__ __forceinline__ float sigm(float x)   { return frcp(1.0f + fexp2(-x * L2E)); }
__device__ __forceinline__ float softp(float x) {
  return (x > 20.0f) ? x : LN2 * flog2(1.0f + fexp2(x * L2E));
}
__device__ __forceinline__ void stnt(float* p, float v) { __builtin_nontemporal_store(v, p); }

// Issue one 32-pixel tile (14 channels) as async DMA global->LDS.
__device__ __forceinline__ void issue_tile_async(const float* __restrict__ raw,
                                                 unsigned gbyte, unsigned lbyte) {
#pragma unroll
  for (int chn = 0; chn < CC; ++chn) {
    asm volatile("global_load_async_to_lds_b32 %0, %1, %2"
                 :
                 : "v"(lbyte + (unsigned)(chn * 32 * 4)),
                   "v"(gbyte + (unsigned)(chn * HWX * 4)),
                   "s"(raw)
                 : "memory");
  }
}

__global__ __launch_bounds__(TPB) void splat_kernel(const float* __restrict__ raw,
                                                    const float* __restrict__ ext,
                                                    const float* __restrict__ intr,
                                                    float* __restrict__ out,
                                                    float* __restrict__ wsf,
                                                    unsigned* __restrict__ wsu) {
  __shared__ float stage[TPB / 32][2][CC][32];   // 28 KB double-buffered stage
  __shared__ float sBig, sSmall;
  __shared__ unsigned cBig, cSmall;

  const int tid = threadIdx.x;
  const int lane = tid & 31;
  const int wid = tid >> 5;

  if (tid == 0) { sBig = 0.0f; sSmall = 0.0f; cBig = 0u; cSmall = 0u; }

  const int bv = blockIdx.x / BLOCKS_PER_BV;
  const int blk_in_bv = blockIdx.x % BLOCKS_PER_BV;
  const int block_base = blk_in_bv * PX_PER_BLOCK;       // pixel base inside bv slice
  const int wave_base = block_base + wid * (PPT * 32);
  const int bi = bv >> 2;                                 // VV == 4

  // camera intrinsics (block-uniform)
  const float f = intr[bi * 9 + 0];
  const float cx = intr[bi * 9 + 2];
  const float cy = intr[bi * 9 + 5];
  const float invf = frcp(f);
  const float mult = 2.0f * invf;                         // sum of inv(K[:2,:2])

  // extrinsics (block-uniform)
  const float R00 = ext[bv * 12 + 0], R01 = ext[bv * 12 + 1], R02 = ext[bv * 12 + 2];
  const float T0  = ext[bv * 12 + 3];
  const float R10 = ext[bv * 12 + 4], R11 = ext[bv * 12 + 5], R12 = ext[bv * 12 + 6];
  const float T1  = ext[bv * 12 + 7];
  const float R20 = ext[bv * 12 + 8], R21 = ext[bv * 12 + 9], R22 = ext[bv * 12 + 10];
  const float T2  = ext[bv * 12 + 11];

  const unsigned bv_byte = (unsigned)bv * (unsigned)(CC * HWX) * 4u;

  float aBig = 0.0f, aSmall = 0.0f;
  unsigned nBig = 0u, nSmall = 0u;

  // prologue: stage tile 0
  issue_tile_async(raw,
                   bv_byte + (unsigned)(wave_base + lane) * 4u,
                   (unsigned)(unsigned long long)&stage[wid][0][0][lane]);

#pragma unroll
  for (int t = 0; t < PPT; ++t) {
    // all LDS reads of the buffer we are about to overwrite must be complete
    asm volatile("s_wait_dscnt 0" ::: "memory");
    if (t + 1 < PPT) {
      issue_tile_async(raw,
                       bv_byte + (unsigned)(wave_base + (t + 1) * 32 + lane) * 4u,
                       (unsigned)(unsigned long long)&stage[wid][(t + 1) & 1][0][lane]);
      asm volatile("s_wait_asynccnt 14" ::: "memory");   // tile t has landed
    } else {
      asm volatile("s_wait_asynccnt 0" ::: "memory");
    }

    float chv[CC];
#pragma unroll
    for (int chn = 0; chn < CC; ++chn) chv[chn] = stage[wid][t & 1][chn][lane];

    const int pb = wave_base + t * 32 + lane;   // pixel index inside bv slice
    const int px = pb & (WW - 1);
    const int py = pb >> 9;
    const size_t p = (size_t)bv * HWX + (size_t)pb;

    // rgb / opacity / depth
    const float g0 = softp(chv[0]);
    const float g1 = softp(chv[1]);
    const float g2 = softp(chv[2]);
    const float depth = frcp(sigm(chv[3]) * 0.99f + 0.01f);   // 1/NEAR-1/FAR=0.99
    const float op = sigm(chv[4]);

    // scales + loss accumulation
    const float s0 = softp(chv[5]) * mult;
    const float s1 = softp(chv[6]) * mult;
    const float s2 = softp(chv[7]) * mult;
#define ACC_SCALE(s)                                                  \
    do {                                                              \
      if ((s) > 0.05f)  { aBig += (s); nBig++; }                      \
      if ((s) < 1e-6f)  { aSmall += (-flog_(s)) * 0.1f; nSmall++; }   \
    } while (0)
    ACC_SCALE(s0); ACC_SCALE(s1); ACC_SCALE(s2);
#undef ACC_SCALE

    // quaternion -> rotation (normalize with +1e-8, eps in two_s)
    float qi = chv[8], qj = chv[9], qk = chv[10], qr = chv[11];
    const float qd = qi * qi + qj * qj + qk * qk + qr * qr;
    const float rn = frcp(fsqrt_(qd) + 1e-8f);
    qi *= rn; qj *= rn; qk *= rn; qr *= rn;
    const float qd2 = qi * qi + qj * qj + qk * qk + qr * qr;
    const float two_s = 2.0f * frcp(qd2 + 1e-8f);
    const float xx = qi * qi, yy = qj * qj, zz = qk * qk;
    const float xy = qi * qj, xz = qi * qk, yz = qj * qk;
    const float xr = qi * qr, yr = qj * qr, zr = qk * qr;
    const float A00 = 1.0f - two_s * (yy + zz), A01 = two_s * (xy - zr), A02 = two_s * (xz + yr);
    const float A10 = two_s * (xy + zr), A11 = 1.0f - two_s * (xx + zz), A12 = two_s * (yz - xr);
    const float A20 = two_s * (xz - yr), A21 = two_s * (yz + xr), A22 = 1.0f - two_s * (xx + yy);

    // ray through (sub)pixel, analytic K^-1
    const float ox = sigm(chv[12]) - 0.5f;
    const float oy = sigm(chv[13]) - 0.5f;
    const float dx = ((float)px + ox - cx) * invf;
    const float dy = ((float)py + oy - cy) * invf;
    const float il = frsq(dx * dx + dy * dy + 1.0f);
    const float d0 = dx * il, d1 = dy * il, d2 = il;
    const float w0 = R00 * d0 + R01 * d1 + R02 * d2;
    const float w1 = R10 * d0 + R11 * d1 + R12 * d2;
    const float w2 = R20 * d0 + R21 * d1 + R22 * d2;
    const float m0 = T0 + w0 * depth;
    const float m1 = T1 + w1 * depth;
    const float m2 = T2 + w2 * depth;

    // M = A diag(s^2) A^T (symmetric)
    const float q0 = s0 * s0, q1 = s1 * s1, q2 = s2 * s2;
    const float M00 = A00 * A00 * q0 + A01 * A01 * q1 + A02 * A02 * q2;
    const float M01 = A00 * A10 * q0 + A01 * A11 * q1 + A02 * A12 * q2;
    const float M02 = A00 * A20 * q0 + A01 * A21 * q1 + A02 * A22 * q2;
    const float M11 = A10 * A10 * q0 + A11 * A11 * q1 + A12 * A12 * q2;
    const float M12 = A10 * A20 * q0 + A11 * A21 * q1 + A12 * A22 * q2;
    const float M22 = A20 * A20 * q0 + A21 * A21 * q1 + A22 * A22 * q2;

    // T = R*M ; cov = T*R^T
    const float t00 = R00 * M00 + R01 * M01 + R02 * M02;
    const float t01 = R00 * M01 + R01 * M11 + R02 * M12;
    const float t02 = R00 * M02 + R01 * M12 + R02 * M22;
    const float t10 = R10 * M00 + R11 * M01 + R12 * M02;
    const float t11 = R10 * M01 + R11 * M11 + R12 * M12;
    const float t12 = R10 * M02 + R11 * M12 + R12 * M22;
    const float t20 = R20 * M00 + R21 * M01 + R22 * M02;
    const float t21 = R20 * M01 + R21 * M11 + R22 * M12;
    const float t22 = R20 * M02 + R21 * M12 + R22 * M22;

    // non-temporal streaming stores (write-once data, keep L2 for input)
    float* om = out + OFF_MEANS + p * 3;
    stnt(om + 0, m0); stnt(om + 1, m1); stnt(om + 2, m2);

    float* oc = out + OFF_COV + p * 9;
    stnt(oc + 0, t00 * R00 + t01 * R01 + t02 * R02);
    stnt(oc + 1, t00 * R10 + t01 * R11 + t02 * R12);
    stnt(oc + 2, t00 * R20 + t01 * R21 + t02 * R22);
    stnt(oc + 3, t10 * R00 + t11 * R01 + t12 * R02);
    stnt(oc + 4, t10 * R10 + t11 * R11 + t12 * R12);
    stnt(oc + 5, t10 * R20 + t11 * R21 + t12 * R22);
    stnt(oc + 6, t20 * R00 + t21 * R01 + t22 * R02);
    stnt(oc + 7, t20 * R10 + t21 * R11 + t22 * R12);
    stnt(oc + 8, t20 * R20 + t21 * R21 + t22 * R22);

    stnt(out + OFF_OP + p, op);

    float* orgb = out + OFF_RGB + p * 3;
    stnt(orgb + 0, g0); stnt(orgb + 1, g1); stnt(orgb + 2, g2);
  }

  // block reduction of loss partials (LDS atomics), then global atomics
  __syncthreads();
  if (nBig)   { atomicAdd(&sBig, aBig);     atomicAdd(&cBig, nBig); }
  if (nSmall) { atomicAdd(&sSmall, aSmall); atomicAdd(&cSmall, nSmall); }
  __syncthreads();
  if (tid == 0) {
    if (cBig)   { atomicAdd(&wsf[0], sBig);   atomicAdd(&wsu[2], cBig); }
    if (cSmall) { atomicAdd(&wsf[1], sSmall); atomicAdd(&wsu[3], cSmall); }
  }
}

__global__ void zero_ws_kernel(float* wsf, unsigned* wsu) {
  wsf[0] = 0.0f; wsf[1] = 0.0f; wsu[2] = 0u; wsu[3] = 0u;
}

__global__ void finalize_kernel(const float* wsf, const unsigned* wsu, float* out) {
  out[OFF_BIG]   = (wsu[2] > 0u) ? (wsf[0] / (float)wsu[2]) : 0.0f;
  out[OFF_SMALL] = (wsu[3] > 0u) ? (wsf[1] / (float)wsu[3]) : 0.0f;
}

extern "C" void kernel_launch(void* const* d_in, const int* in_sizes, int n_in,
                              void* d_out, int out_size, void* d_ws, size_t ws_size,
                              hipStream_t stream) {
  (void)in_sizes; (void)n_in; (void)out_size; (void)ws_size;
  const float* raw  = (const float*)d_in[0];
  const float* ext  = (const float*)d_in[1];
  const float* intr = (const float*)d_in[2];
  float* out = (float*)d_out;
  float* wsf = (float*)d_ws;
  unsigned* wsu = (unsigned*)d_ws;

  zero_ws_kernel<<<1, 1, 0, stream>>>(wsf, wsu);
  splat_kernel<<<NBLOCKS, TPB, 0, stream>>>(raw, ext, intr, out, wsf, wsu);
  finalize_kernel<<<1, 1, 0, stream>>>(wsf, wsu, out);
}